// Pooling_6476810682592
// MI455X (gfx1250) — compile-verified
//
#include <hip/hip_runtime.h>

typedef float v2f __attribute__((ext_vector_type(2)));
typedef float v8f __attribute__((ext_vector_type(8)));

#define BS       4
#define NV       4096
#define NC       256
#define POOL     1024
#define KNN      16
#define NTILES   (NV / 16)

// ---------------------------------------------------------------------------
// Kernel 1: 16-NN for the 1024 sampled query rows per batch.
// One wave (32 threads) per (batch, 16-query tile). Gram tiles via
// V_WMMA_F32_16X16X4_F32 (K=3 padded to 4); per-row top-16 kept in registers.
// ---------------------------------------------------------------------------
__global__ __launch_bounds__(32) void knn_wmma_kernel(
    const float* __restrict__ V,     // [BS, NV, 3]
    const int*   __restrict__ sidx,  // [POOL]
    int*         __restrict__ nbr)   // [BS, POOL, KNN]
{
    __shared__ float qqs[16];
    __shared__ float dls[16 * 17];   // stride 17: conflict-free row reads

    const int wg   = blockIdx.x;      // 0..BS*64-1
    const int b    = wg >> 6;
    const int tile = wg & 63;
    const int lane = threadIdx.x;     // 0..31
    const int L    = lane & 15;
    const int g    = lane >> 4;       // 0: lanes 0-15, 1: lanes 16-31

    // ---- A matrix: 16 sampled query vertices, K = (x,y | z,0) ------------
    const int m  = tile * 16 + L;          // pooled row id 0..1023
    const int sq = sidx[m];                // original vertex id of query
    const float* qp = V + ((size_t)b * NV + (size_t)sq) * 3;
    const float qx = qp[0], qy = qp[1], qz = qp[2];

    v2f a;
    a.x = g ? qz : qx;
    a.y = g ? 0.0f : qy;

    if (g == 0) qqs[L] = qx * qx + qy * qy + qz * qz;
    __syncthreads();

    float qqv[8];                           // |q|^2 for rows this half owns
#pragma unroll
    for (int r = 0; r < 8; ++r) qqv[r] = qqs[r + 8 * g];
    __syncthreads();

    // ---- per-lane sorted top-16 (ascending), register resident -----------
    float lst[KNN];
    int   il [KNN];
#pragma unroll
    for (int p = 0; p < KNN; ++p) { lst[p] = 3.402823466e38f; il[p] = 0; }

    const float* VB = V + (size_t)b * NV * 3;

    for (int tc = 0; tc < NTILES; ++tc) {
        const int n0 = tc * 16;

        // B matrix: 16 candidate vertices, K = (x,y | z,0)
        const float* cp = VB + (size_t)(n0 + L) * 3;
        const float cx = cp[0], cy = cp[1], cz = cp[2];
        const float cc = cx * cx + cy * cy + cz * cz;

        v2f bv;
        bv.x = g ? cz : cx;
        bv.y = g ? 0.0f : cy;

        v8f acc = {0.f, 0.f, 0.f, 0.f, 0.f, 0.f, 0.f, 0.f};
        acc = __builtin_amdgcn_wmma_f32_16x16x4_f32(
                  false, a, false, bv, (short)0, acc, false, false);

        // distance tile -> LDS (row r+8g, col L), padded stride 17
#pragma unroll
        for (int r = 0; r < 8; ++r) {
            const float dist = qqv[r] + cc - 2.0f * acc[r];
            dls[(r + 8 * g) * 17 + L] = dist;
        }
        __syncthreads();

        // lanes 0-15: each owns one query row; scan the 16 candidates.
        // Preload the whole row -> registers (one grouped LDS burst, single
        // dscnt wait), then a fully unrolled scan so everything stays in
        // registers with constant indices.
        if (g == 0) {
            float dv[16];
#pragma unroll
            for (int j = 0; j < 16; ++j) dv[j] = dls[L * 17 + j];

#pragma unroll
            for (int j = 0; j < 16; ++j) {
                const float dj = dv[j];
                const int   gj = n0 + j;
                if (gj != sq && dj < lst[KNN - 1]) {
                    // branch-free unrolled insertion (constant indices only)
#pragma unroll
                    for (int p = KNN - 1; p > 0; --p) {
                        const bool shift = dj < lst[p - 1];
                        const bool here  = (!shift) && (dj < lst[p]);
                        lst[p] = shift ? lst[p - 1] : (here ? dj : lst[p]);
                        il [p] = shift ? il [p - 1] : (here ? gj : il [p]);
                    }
                    if (dj < lst[0]) { lst[0] = dj; il[0] = gj; }
                }
            }
        }
        __syncthreads();   // protect dls before next tile overwrites it
    }

    if (g == 0) {
        int* out = nbr + ((size_t)b * POOL + (size_t)m) * KNN;
#pragma unroll
        for (int k = 0; k < KNN; ++k) out[k] = il[k];
    }
}

// ---------------------------------------------------------------------------
// Kernel 2: gather 16 neighbor feature rows, per-channel max, plus the
// sampled-vertex gather. 1 block per pooled row, 1 thread per channel.
// Feature rows stream out of L2 (16 MB map << 192 MB L2), fully coalesced.
// ---------------------------------------------------------------------------
__global__ __launch_bounds__(NC) void pool_gather_kernel(
    const float* __restrict__ V,     // [BS, NV, 3]
    const float* __restrict__ F,     // [BS, NV, NC]
    const int*   __restrict__ sidx,  // [POOL]
    const int*   __restrict__ nbr,   // [BS, POOL, KNN]
    float*       __restrict__ outV,  // [BS, POOL, 3]
    float*       __restrict__ outF)  // [BS, POOL, NC]
{
    const int row = blockIdx.x;            // 0 .. BS*POOL-1
    const int b   = row >> 10;
    const int m   = row & (POOL - 1);
    const int t   = threadIdx.x;           // channel

    const int* nb = nbr + (size_t)row * KNN;   // uniform -> scalar loads
    const float* FB = F + (size_t)b * NV * NC;

    float mx = -3.402823466e38f;
#pragma unroll
    for (int k = 0; k < KNN; ++k) {
        const int j = nb[k];
        mx = fmaxf(mx, FB[(size_t)j * NC + t]);
    }
    outF[(size_t)row * NC + t] = mx;

    if (t < 3) {
        const int s = sidx[m];
        outV[(size_t)row * 3 + t] = V[((size_t)b * NV + (size_t)s) * 3 + t];
    }
}

// ---------------------------------------------------------------------------
extern "C" void kernel_launch(void* const* d_in, const int* in_sizes, int n_in,
                              void* d_out, int out_size, void* d_ws, size_t ws_size,
                              hipStream_t stream)
{
    const float* V    = (const float*)d_in[0];   // vertices  [4,4096,3]
    const float* F    = (const float*)d_in[1];   // features  [4,4096,256]
    const int*   sidx = (const int*)  d_in[2];   // sample_idx [1024]

    float* outV = (float*)d_out;                 // [4,1024,3]   = 12288 floats
    float* outF = outV + (size_t)BS * POOL * 3;  // [4,1024,256]

    int* nbr = (int*)d_ws;                       // [4,1024,16] = 256 KB

    knn_wmma_kernel<<<BS * (POOL / 16), 32, 0, stream>>>(V, sidx, nbr);
    pool_gather_kernel<<<BS * POOL, NC, 0, stream>>>(V, F, sidx, nbr, outV, outF);
}